// GraphGenerativeModel_30477087932727
// MI455X (gfx1250) — compile-verified
//
#include <hip/hip_runtime.h>
#include <hip/hip_bf16.h>
#include <math.h>

// ---------------------------------------------------------------------------
// VGAE forward for MI455X (gfx1250, wave32).
// Dense GEMMs -> V_WMMA_F32_16X16X4_F32 (fp32 matrix path).
// LDS staging  -> TDM tensor_load_to_lds (decoder) and
//                 global_load_async_to_lds_b128 (encoder GEMM).
// ---------------------------------------------------------------------------

typedef __attribute__((ext_vector_type(2))) float    v2f;
typedef __attribute__((ext_vector_type(8))) float    v8f;
typedef __attribute__((ext_vector_type(4))) unsigned v4u;
typedef __attribute__((ext_vector_type(8))) int      v8i;
typedef __attribute__((ext_vector_type(4))) int      v4i;

#define NN     10000   // nodes (divisible by 16)
#define EE     320000  // edges
#define DIN    512
#define DZ     128

__device__ __forceinline__ v8f wmma_f32x4(v2f a, v2f b, v8f c) {
  // 8 args: (neg_a, A, neg_b, B, c_mod, C, reuse_a, reuse_b)
  return __builtin_amdgcn_wmma_f32_16x16x4_f32(
      false, a, false, b, (short)0, c, false, false);
}

// low 32 bits of a generic (flat) pointer into the LDS aperture are the
// hardware LDS byte address (ISA 10.2)
__device__ __forceinline__ unsigned lds_byte_addr(const void* p) {
  return (unsigned)(unsigned long long)(uintptr_t)p;
}

// ---------------------------------------------------------------------------
// TDM: DMA a 16x128 f32 tile (row stride 128 floats) from global into LDS.
// Descriptor per CDNA5 ISA ch.8 (D# groups 0/1; 2D tensor -> groups 2/3 zero).
// Issue from ONE wave only; completion via S_WAIT_TENSORCNT in that wave.
// ---------------------------------------------------------------------------
__device__ __forceinline__ void tdm_load_tile_16x128(unsigned lds_addr,
                                                     const float* gsrc) {
  const unsigned long long ga = (unsigned long long)(uintptr_t)gsrc;
  v4u g0;
  g0.x = 1u;                                   // count=1, user mode, no gather
  g0.y = lds_addr;                             // [63:32]  lds_addr
  g0.z = (unsigned)(ga & 0xFFFFFFFFu);         // [95:64]  global_addr lo
  g0.w = (unsigned)((ga >> 32) & 0x01FFFFFFu)  // [120:96] global_addr hi
         | (2u << 30);                         // [127:126] type=2 (image)
  v8i g1;
  g1[0] = (int)(2u << 16);                     // data_size=2 (4B), wg_mask=0
  g1[1] = (int)(128u << 16);                   // tensor_dim0 lo16 = 128
  g1[2] = (int)(16u << 16);                    // dim0 hi=0 | tensor_dim1 lo16=16
  g1[3] = (int)(128u << 16);                   // dim1 hi=0 | tile_dim0=128
  g1[4] = (int)(16u);                          // tile_dim1=16, tile_dim2=0
  g1[5] = (int)(128u);                         // tensor_dim0_stride = 128
  g1[6] = 0;                                   // stride0 hi | stride1 lo
  g1[7] = 0;                                   // stride1 hi
  v4i g2 = {0, 0, 0, 0};                       // group 2 (unused, 2D tensor)
  v4i g3 = {0, 0, 0, 0};                       // group 3 (unused, 2D tensor)
  v8i g4 = {0, 0, 0, 0, 0, 0, 0, 0};           // trailing group (clang-23 sig)
  __builtin_amdgcn_tensor_load_to_lds(g0, g1, g2, g3, g4, 0);
}

// ---------------------------------------------------------------------------
// Generic  out[M x 128] = act(A)[M x K] @ W[128 x K]^T + bias
// One block = 16 output rows x all 128 cols; 8 waves, one 16x16 tile each.
// !PRELU: A tile staged via async global->LDS DMA (ASYNCcnt path).
//  PRELU: manual stage (activation applied on the way in).
// ---------------------------------------------------------------------------
template <int K, bool PRELU>
__global__ void __launch_bounds__(256)
gemm_wt_kernel(const float* __restrict__ A, const float* __restrict__ W,
               const float* __restrict__ bias, const float* __restrict__ alpha,
               float* __restrict__ out) {
  __shared__ float As[16 * K];

  const int m0 = blockIdx.x * 16;

  if constexpr (!PRELU) {
    // async copy: 16*K floats in 16B chunks, coalesced, no VGPR staging
    const unsigned lbase = lds_byte_addr(As);
    const float* gsrc = A + (size_t)m0 * K;
    for (int c = threadIdx.x * 4; c < 16 * K; c += 256 * 4) {
      asm volatile("global_load_async_to_lds_b128 %0, %1, off"
                   :
                   : "v"(lbase + (unsigned)c * 4u), "v"(gsrc + c)
                   : "memory");
    }
    asm volatile("s_wait_asynccnt 0x0" ::: "memory");
  } else {
    const float a_neg = alpha[0];
    for (int idx = threadIdx.x; idx < 16 * K; idx += 256) {
      float v = A[(size_t)m0 * K + idx];
      As[idx] = (v >= 0.0f) ? v : a_neg * v;
    }
  }
  __syncthreads();

  const int wave   = threadIdx.x >> 5;
  const int lane   = threadIdx.x & 31;
  const int m_half = lane >> 4;
  const int mn     = lane & 15;
  const int n      = wave * 16 + mn;

  v8f c;
  const float bv = bias[n];
#pragma unroll
  for (int v = 0; v < 8; ++v) c[v] = bv;

  const float* wrow = W + (size_t)n * K;   // B[k][n] = W[n][k]
  __builtin_prefetch(wrow, 0, 0);

#pragma unroll 4
  for (int kk = 0; kk < K; kk += 4) {
    const int ka = kk + 2 * m_half;
    v2f a = *(const v2f*)&As[mn * K + ka];
    v2f b = *(const v2f*)&wrow[ka];
    c = wmma_f32x4(a, b, c);
  }

#pragma unroll
  for (int v = 0; v < 8; ++v) {
    const int m = m0 + v + 8 * m_half;
    out[(size_t)m * DZ + n] = c[v];
  }
}

// ---------------------------------------------------------------------------
// zero fp32 buffer (float4 stores)
// ---------------------------------------------------------------------------
__global__ void __launch_bounds__(256)
zero_kernel(float4* __restrict__ p, int n4) {
  int i = blockIdx.x * 256 + threadIdx.x;
  if (i < n4) p[i] = make_float4(0.f, 0.f, 0.f, 0.f);
}

// ---------------------------------------------------------------------------
// SpMM scatter: agg[row[e], :] += val[e] * lin[col[e], :]
// one wave per edge; each lane owns 4 of 128 dims (float4 load, 4 atomics)
// ---------------------------------------------------------------------------
__global__ void __launch_bounds__(256)
spmm_kernel(const int* __restrict__ erow, const int* __restrict__ ecol,
            const float* __restrict__ eval, const float* __restrict__ lin,
            float* __restrict__ agg) {
  const int e = (blockIdx.x * 256 + threadIdx.x) >> 5;
  if (e >= EE) return;
  const int lane = threadIdx.x & 31;
  const int r = erow[e];
  const int cnode = ecol[e];
  const float v = eval[e];
  const float4 m = *(const float4*)&lin[(size_t)cnode * DZ + lane * 4];
  float* dst = &agg[(size_t)r * DZ + lane * 4];
  atomicAdd(dst + 0, v * m.x);
  atomicAdd(dst + 1, v * m.y);
  atomicAdd(dst + 2, v * m.z);
  atomicAdd(dst + 3, v * m.w);
}

// ---------------------------------------------------------------------------
// Latent heads: h = PReLU(agg1); z_mean = h@Wm^T+bm; z_std = softplus(h@Ws^T+bs)
// z = z_mean + noise * z_std.  512 threads: waves 0-7 -> mean, 8-15 -> std.
// ---------------------------------------------------------------------------
__global__ void __launch_bounds__(512)
latent_kernel(const float* __restrict__ agg1, const float* __restrict__ alpha1,
              const float* __restrict__ Wm, const float* __restrict__ bm,
              const float* __restrict__ Ws, const float* __restrict__ bs,
              const float* __restrict__ noise,
              float* __restrict__ zmean_out, float* __restrict__ zstd_out,
              float* __restrict__ z_out) {
  __shared__ float Hs[16 * DZ];
  __shared__ float Ms[16 * DZ];
  __shared__ float Ss[16 * DZ];

  const int m0 = blockIdx.x * 16;
  const float a1 = alpha1[0];
  for (int idx = threadIdx.x; idx < 16 * DZ; idx += 512) {
    float v = agg1[(size_t)m0 * DZ + idx];
    Hs[idx] = (v >= 0.0f) ? v : a1 * v;
  }
  __syncthreads();

  const int wave   = threadIdx.x >> 5;
  const int lane   = threadIdx.x & 31;
  const bool sstd  = (wave >= 8);
  const int n0     = (wave & 7) * 16;
  const int m_half = lane >> 4;
  const int mn     = lane & 15;
  const int n      = n0 + mn;

  const float* W = sstd ? Ws : Wm;
  const float* b = sstd ? bs : bm;

  v8f c;
  const float bv = b[n];
#pragma unroll
  for (int v = 0; v < 8; ++v) c[v] = bv;

  const float* wrow = W + (size_t)n * DZ;
  __builtin_prefetch(wrow, 0, 0);
#pragma unroll 4
  for (int kk = 0; kk < DZ; kk += 4) {
    const int ka = kk + 2 * m_half;
    v2f a  = *(const v2f*)&Hs[mn * DZ + ka];
    v2f bb = *(const v2f*)&wrow[ka];
    c = wmma_f32x4(a, bb, c);
  }

  float* dst = sstd ? Ss : Ms;
#pragma unroll
  for (int v = 0; v < 8; ++v) dst[(v + 8 * m_half) * DZ + n] = c[v];
  __syncthreads();

  for (int idx = threadIdx.x; idx < 16 * DZ; idx += 512) {
    const float mean = Ms[idx];
    const float sl   = Ss[idx];
    const float sp = fmaxf(sl, 0.0f) + log1pf(__expf(-fabsf(sl)));
    const float zv = mean + noise[(size_t)m0 * DZ + idx] * sp;
    zmean_out[(size_t)m0 * DZ + idx] = mean;
    zstd_out [(size_t)m0 * DZ + idx] = sp;
    z_out    [(size_t)m0 * DZ + idx] = zv;
  }
}

// ---------------------------------------------------------------------------
// Decoder: a_probs = sigmoid(z @ z^T)   [10000 x 10000]
// A tile (16x128 of z) delivered by the Tensor Data Mover straight into LDS;
// B rows of z read from L2 (z is 5 MB, fully L2 resident).
// ---------------------------------------------------------------------------
__global__ void __launch_bounds__(256)
decoder_kernel(const float* __restrict__ z, float* __restrict__ out) {
  __shared__ float As[16 * DZ];
  const int m0   = blockIdx.x * 16;
  const int wave = threadIdx.x >> 5;

  if (wave == 0) {   // one TDM descriptor per workgroup (TENSORcnt is per-wave)
    tdm_load_tile_16x128(lds_byte_addr(As), z + (size_t)m0 * DZ);
    __builtin_amdgcn_s_wait_tensorcnt(0);
  }
  __syncthreads();   // publish the DMA'd tile to all 8 waves

  const int ntile = blockIdx.y * 8 + wave;
  if (ntile < NN / 16) {   // wave-uniform guard: EXEC stays all-1s inside
    const int lane   = threadIdx.x & 31;
    const int m_half = lane >> 4;
    const int mn     = lane & 15;
    const int n      = ntile * 16 + mn;

    v8f c;
#pragma unroll
    for (int v = 0; v < 8; ++v) c[v] = 0.0f;

    const float* zb = z + (size_t)n * DZ;   // B[k][n] = z[n][k]
    __builtin_prefetch(zb, 0, 0);
#pragma unroll 8
    for (int kk = 0; kk < DZ; kk += 4) {
      const int ka = kk + 2 * m_half;
      v2f a = *(const v2f*)&As[mn * DZ + ka];
      v2f b = *(const v2f*)&zb[ka];
      c = wmma_f32x4(a, b, c);
    }

#pragma unroll
    for (int v = 0; v < 8; ++v) {
      const int m = m0 + v + 8 * m_half;
      const float p = 1.0f / (1.0f + __expf(-c[v]));
      out[(size_t)m * NN + n] = p;
    }
  }
}

// ---------------------------------------------------------------------------
// host launcher
// ---------------------------------------------------------------------------
extern "C" void kernel_launch(void* const* d_in, const int* in_sizes, int n_in,
                              void* d_out, int out_size, void* d_ws, size_t ws_size,
                              hipStream_t stream) {
  const float* x    = (const float*)d_in[0];
  const int*   erow = (const int*)  d_in[1];
  const int*   ecol = (const int*)  d_in[2];
  const float* eval = (const float*)d_in[3];
  const float* nois = (const float*)d_in[4];
  const float* W0   = (const float*)d_in[5];
  const float* b0   = (const float*)d_in[6];
  const float* a0   = (const float*)d_in[7];
  const float* W1   = (const float*)d_in[8];
  const float* b1   = (const float*)d_in[9];
  const float* a1   = (const float*)d_in[10];
  const float* Wm   = (const float*)d_in[11];
  const float* bm   = (const float*)d_in[12];
  const float* Ws   = (const float*)d_in[13];
  const float* bs   = (const float*)d_in[14];

  float* aprobs = (float*)d_out;                       // [N*N]
  float* zmean  = aprobs + (size_t)NN * NN;            // [N*DZ]
  float* zstd   = zmean + (size_t)NN * DZ;             // [N*DZ]

  const size_t buf = (size_t)NN * DZ;                  // 1.28M floats each
  float* lin = (float*)d_ws;          // ws0: linear outputs
  float* agg = lin + buf;             // ws1: spmm accumulators
  float* z   = agg + buf;             // ws2: latent z
  (void)in_sizes; (void)n_in; (void)out_size; (void)ws_size;

  const int mblocks = NN / 16;                         // 625
  const int zero_b  = (int)((buf / 4 + 255) / 256);    // float4 zeroing
  const int spmm_b  = (EE * 32 + 255) / 256;           // one wave per edge

  // layer 0: lin = x @ W0^T + b0   (async-to-LDS staging)
  gemm_wt_kernel<DIN, false><<<mblocks, 256, 0, stream>>>(x, W0, b0, a0, lin);
  // agg = spmm(adj, lin)
  zero_kernel<<<zero_b, 256, 0, stream>>>((float4*)agg, (int)(buf / 4));
  spmm_kernel<<<spmm_b, 256, 0, stream>>>(erow, ecol, eval, lin, agg);
  // layer 1: lin = PReLU(agg) @ W1^T + b1   (PReLU fused into A-tile stage)
  gemm_wt_kernel<DZ, true><<<mblocks, 256, 0, stream>>>(agg, W1, b1, a0, lin);
  // agg = spmm(adj, lin)
  zero_kernel<<<zero_b, 256, 0, stream>>>((float4*)agg, (int)(buf / 4));
  spmm_kernel<<<spmm_b, 256, 0, stream>>>(erow, ecol, eval, lin, agg);
  // latent heads + reparameterize (PReLU(a1) fused)
  latent_kernel<<<mblocks, 512, 0, stream>>>(agg, a1, Wm, bm, Ws, bs, nois,
                                             zmean, zstd, z);
  // dense decoder: sigmoid(z @ z^T), TDM-staged A tiles
  dim3 dgrid(mblocks, (mblocks + 7) / 8);              // 625 x 79
  decoder_kernel<<<dgrid, 256, 0, stream>>>(z, aprobs);
}